// ASTEnc_80616536146350
// MI455X (gfx1250) — compile-verified
//
#include <hip/hip_runtime.h>

#define NN 50000
#define DD 128
#define LL 3
#define EPSI 1e-5f

typedef float v2f __attribute__((ext_vector_type(2)));
typedef float v8f __attribute__((ext_vector_type(8)));

// ---------------------------------------------------------------- zero agg
__global__ void zero_kernel(float* __restrict__ p, int n4) {
    int i = blockIdx.x * blockDim.x + threadIdx.x;
    if (i < n4) ((float4*)p)[i] = make_float4(0.f, 0.f, 0.f, 0.f);
}

// ------------------------------------------- embedding + LayerNorm (1 wave/row)
__global__ __launch_bounds__(256)
void embed_ln_kernel(const float* __restrict__ node_emb,
                     const int*   __restrict__ pos,
                     const float* __restrict__ pos_table,
                     const float* __restrict__ g,
                     const float* __restrict__ b,
                     float* __restrict__ x) {
    const int wave = threadIdx.x >> 5;
    const int lane = threadIdx.x & 31;
    const int n = blockIdx.x * 8 + wave;          // N = 6250*8 exactly

    const float4 e  = ((const float4*)(node_emb + (size_t)n * DD))[lane];
    const int    p  = pos[n];
    const float4 pe = ((const float4*)(pos_table + (size_t)p * DD))[lane];
    const float sc = 11.313708498984761f;          // sqrt(128)

    float v[4] = { e.x*sc + pe.x, e.y*sc + pe.y, e.z*sc + pe.z, e.w*sc + pe.w };
    float s = 0.f, s2 = 0.f;
    #pragma unroll
    for (int j = 0; j < 4; ++j) { s += v[j]; s2 += v[j]*v[j]; }
    #pragma unroll
    for (int m = 16; m; m >>= 1) {                 // wave32 reduction
        s  += __shfl_xor(s,  m, 32);
        s2 += __shfl_xor(s2, m, 32);
    }
    const float mean = s * (1.f / DD);
    const float var  = s2 * (1.f / DD) - mean * mean;
    const float rs   = rsqrtf(var + EPSI);

    const float4 gv = ((const float4*)g)[lane];
    const float4 bv = ((const float4*)b)[lane];
    float4 o;
    o.x = (v[0]-mean)*rs*gv.x + bv.x;
    o.y = (v[1]-mean)*rs*gv.y + bv.y;
    o.z = (v[2]-mean)*rs*gv.z + bv.z;
    o.w = (v[3]-mean)*rs*gv.w + bv.w;
    ((float4*)(x + (size_t)n * DD))[lane] = o;
}

// ---------------------------------------- scatter-add: one wave per edge
__global__ __launch_bounds__(256)
void scatter_kernel(const int* __restrict__ edge,    // [2,E]: src then dst
                    const float* __restrict__ x,
                    float* __restrict__ agg, int E) {
    int t = blockIdx.x * blockDim.x + threadIdx.x;
    int e = t >> 5;
    int lane = t & 31;
    if (e >= E) return;
    const int src = edge[e];
    const int dst = edge[E + e];
    const float4 v = ((const float4*)(x + (size_t)src * DD))[lane];
    float* a = agg + (size_t)dst * DD + lane * 4;
    unsafeAtomicAdd(a + 0, v.x);
    unsafeAtomicAdd(a + 1, v.y);
    unsafeAtomicAdd(a + 2, v.z);
    unsafeAtomicAdd(a + 3, v.w);
}

// ---------- fused SAGE layer: h = relu(agg@Wl + bl + x@Wr); x = LN(h + x)
// block: 256 thr = 8 waves; one 16-row strip per block; wave w -> cols [16w,16w+16)
__global__ __launch_bounds__(256)
void sage_layer_kernel(const float* __restrict__ xin,
                       const float* __restrict__ agg,
                       const float* __restrict__ Wl,
                       const float* __restrict__ bl,
                       const float* __restrict__ Wr,
                       const float* __restrict__ g,
                       const float* __restrict__ bb,
                       float* __restrict__ xout) {
    __shared__ float sh_x[16 * DD];
    __shared__ float sh_a[16 * DD];   // agg tile, then reused for h + x

    const int tid  = threadIdx.x;
    const int wave = tid >> 5;
    const int lane = tid & 31;
    const int row0 = blockIdx.x * 16;              // N = 3125*16 exactly

    // stage 16x128 activation tiles (2 float4 per thread per tile)
    {
        const float4* gx = (const float4*)(xin + (size_t)row0 * DD);
        const float4* ga = (const float4*)(agg + (size_t)row0 * DD);
        float4* sx = (float4*)sh_x;
        float4* sa = (float4*)sh_a;
        sx[tid]       = gx[tid];
        sx[tid + 256] = gx[tid + 256];
        sa[tid]       = ga[tid];
        sa[tid + 256] = ga[tid + 256];
    }
    __syncthreads();

    const int half = lane >> 4;                    // 0 | 1
    const int l16  = lane & 15;
    const int col  = wave * 16 + l16;              // this lane's output column

    v8f acc = {};
    #pragma unroll 4
    for (int k0 = 0; k0 < DD; k0 += 4) {
        const int ka = k0 + 2 * half;              // K index for .x; .y = ka+1
        v2f a0, a1, b0, b1;
        a0.x = sh_a[l16 * DD + ka];     a0.y = sh_a[l16 * DD + ka + 1];
        a1.x = sh_x[l16 * DD + ka];     a1.y = sh_x[l16 * DD + ka + 1];
        b0.x = Wl[(size_t)ka * DD + col];       b0.y = Wl[(size_t)(ka + 1) * DD + col];
        b1.x = Wr[(size_t)ka * DD + col];       b1.y = Wr[(size_t)(ka + 1) * DD + col];
        acc = __builtin_amdgcn_wmma_f32_16x16x4_f32(false, a0, false, b0,
                                                    (short)0, acc, false, false);
        acc = __builtin_amdgcn_wmma_f32_16x16x4_f32(false, a1, false, b1,
                                                    (short)0, acc, false, false);
    }

    const float bias = bl[col];
    __syncthreads();                               // all waves done reading sh_a
    #pragma unroll
    for (int r = 0; r < 8; ++r) {                  // D-matrix: M = r + 8*half
        const int row = r + 8 * half;
        float h = fmaxf(acc[r] + bias, 0.f);       // bias + relu
        sh_a[row * DD + col] = h + sh_x[row * DD + col];   // residual
    }
    __syncthreads();

    // LayerNorm: 16 threads per row, 8 elems per thread
    const int row = tid >> 4;
    const int seg = tid & 15;
    const float* hr = sh_a + row * DD + seg * 8;
    float vals[8];
    float s = 0.f, s2 = 0.f;
    #pragma unroll
    for (int j = 0; j < 8; ++j) { float v = hr[j]; vals[j] = v; s += v; s2 += v*v; }
    #pragma unroll
    for (int m = 8; m; m >>= 1) {                  // stays within 16-lane group
        s  += __shfl_xor(s,  m, 32);
        s2 += __shfl_xor(s2, m, 32);
    }
    const float mean = s * (1.f / DD);
    const float var  = s2 * (1.f / DD) - mean * mean;
    const float rs   = rsqrtf(var + EPSI);

    float* orow = xout + (size_t)(row0 + row) * DD + seg * 8;
    const float* gp = g  + seg * 8;
    const float* bp = bb + seg * 8;
    #pragma unroll
    for (int j = 0; j < 8; ++j)
        orow[j] = (vals[j] - mean) * rs * gp[j] + bp[j];
}

// ---------------------------------------------------------------- launch
extern "C" void kernel_launch(void* const* d_in, const int* in_sizes, int n_in,
                              void* d_out, int out_size, void* d_ws, size_t ws_size,
                              hipStream_t stream) {
    const float* node_emb  = (const float*)d_in[0];
    const int*   pos       = (const int*)  d_in[1];
    const int*   edge      = (const int*)  d_in[2];
    const float* pos_table = (const float*)d_in[3];
    const float* Wl        = (const float*)d_in[4];
    const float* bl        = (const float*)d_in[5];
    const float* Wr        = (const float*)d_in[6];
    const float* emb_g     = (const float*)d_in[7];
    const float* emb_b     = (const float*)d_in[8];
    const float* hid_g     = (const float*)d_in[9];
    const float* hid_b     = (const float*)d_in[10];
    const int E = in_sizes[2] / 2;

    float* x   = (float*)d_ws;                    // [N, D]
    float* agg = x + (size_t)NN * DD;             // [N, D]
    float* out = (float*)d_out;

    embed_ln_kernel<<<NN / 8, 256, 0, stream>>>(node_emb, pos, pos_table,
                                                emb_g, emb_b, x);
    for (int i = 0; i < LL; ++i) {
        zero_kernel<<<(NN * DD / 4 + 255) / 256, 256, 0, stream>>>(agg, NN * DD / 4);
        const int nwork = E * 32;                 // one wave per edge
        scatter_kernel<<<(nwork + 255) / 256, 256, 0, stream>>>(edge, x, agg, E);
        float* dst = (i == LL - 1) ? out : x;
        sage_layer_kernel<<<NN / 16, 256, 0, stream>>>(
            x, agg,
            Wl + (size_t)i * DD * DD, bl + (size_t)i * DD,
            Wr + (size_t)i * DD * DD,
            hid_g + (size_t)i * DD, hid_b + (size_t)i * DD, dst);
    }
}